// Model_10943576670968
// MI455X (gfx1250) — compile-verified
//
#include <hip/hip_runtime.h>

// ---------------------------------------------------------------------------
// RGCN 2-hop propagation for MI455X (gfx1250, wave32).
//   h2  = ent_emb[final_indices]                      (virtual, never stored)
//   h1  = tanh(ent_emb[ent_now1]@Ws + b + norm(segsum(h2[src]*rel[r]))@Wn)
//   usr =      user_emb         @Ws + b + norm(segsum(h1[src]*rel[r]))@Wn
//   unr = tanh(ent_emb[unreach] @Ws + b)
//   out = [usr ; concat(h1,h2,unr)[order]]
//
// GEMMs: V_WMMA_F32_16X16X4_F32 (exact fp32), 2 row-tiles per wave, all B
// fragments for a K-step loaded in one clause before the WMMA burst.
// All variant selection is via templates -> no per-lane EXEC masking around
// WMMA (EXEC must be all-1s for WMMA).
// Scatter: f32 atomics; tables + agg buffers are L2-resident (192MB L2).
// ---------------------------------------------------------------------------

typedef __attribute__((ext_vector_type(2))) float v2f;
typedef __attribute__((ext_vector_type(8))) float v8f;

constexpr int E_  = 200000, U_ = 50000, D_ = 128;
constexpr int N2_ = 100000, N1_ = 80000, NU_ = 20000;
constexpr int M1_ = 1600000, M0_ = 800000;

__global__ void zero_kernel(float* __restrict__ p, long n) {
    long i = (long)blockIdx.x * blockDim.x + threadIdx.x;
    long stride = (long)gridDim.x * blockDim.x;
    for (; i < n; i += stride) p[i] = 0.0f;
}

// One wave32 per edge: msg = tail[GATHER?gidx[src]:src] * rel_emb[rel],
// atomically accumulated into agg[dst]; lane 0 counts degree.
template<bool GATHER>
__global__ __launch_bounds__(256)
void edge_scatter_kernel(const float* __restrict__ tail,
                         const int*   __restrict__ gidx,
                         const int*   __restrict__ src,
                         const int*   __restrict__ dst,
                         const int*   __restrict__ rel,
                         const float* __restrict__ rel_emb,
                         float*       __restrict__ agg,
                         float*       __restrict__ deg,
                         int M)
{
    int edge = blockIdx.x * (blockDim.x >> 5) + (threadIdx.x >> 5);
    if (edge >= M) return;
    int lane = threadIdx.x & 31;

    int s = src[edge];
    int d = dst[edge];
    int r = rel[edge];
    long trow = (long)(GATHER ? gidx[s] : s) * D_;

    const float4* t4 = (const float4*)(tail + trow);
    const float4* r4 = (const float4*)(rel_emb + (long)r * D_);
    float4 tv = t4[lane];   // 512B row covered by 32 lanes x float4
    float4 rv = r4[lane];

    float* arow = agg + (long)d * D_ + lane * 4;
    atomicAdd(arow + 0, tv.x * rv.x);
    atomicAdd(arow + 1, tv.y * rv.y);
    atomicAdd(arow + 2, tv.z * rv.z);
    atomicAdd(arow + 3, tv.w * rv.w);
    if (lane == 0) atomicAdd(deg + d, 1.0f);
}

// One wave32 per 32-row strip (2 x 16-row WMMA tiles) of the 128-wide output:
//   out[32,128] = act( X[32,128]@W_self + b + (agg/deg)[32,128]@W_nei )
// A layout (ISA 7.12.2, 32-bit A 16x4): lane m (0..15) VGPR{0,1}=K{kb,kb+1};
// lane m+16 = K{kb+2,kb+3} -> per-lane float2 at row[kb + 2*hi].
// B layout: VGPR0 = W[kb+2hi][n], VGPR1 = W[kb+2hi+1][n], n = 16c+m.
template<bool GATHER, bool USE_AGG, bool ACT>
__global__ __launch_bounds__(256)
void rgcn_node_kernel(const float* __restrict__ X,
                      const int*   __restrict__ gidx,
                      const float* __restrict__ agg,
                      const float* __restrict__ deg,
                      const float* __restrict__ W_self,
                      const float* __restrict__ W_nei,
                      const float* __restrict__ bias,
                      float*       __restrict__ out,
                      int n_rows)
{
    int wid  = blockIdx.x * (blockDim.x >> 5) + (threadIdx.x >> 5);
    int row0 = wid * 32;
    if (row0 >= n_rows) return;            // wave-uniform exit: EXEC stays full

    int lane = threadIdx.x & 31;
    int m    = lane & 15;
    int hi   = lane >> 4;

    int rA = row0 + m;
    int rBraw = row0 + 16 + m;
    int rB = (rBraw < n_rows) ? rBraw : rA;   // clamp loads (select, no branch)

    const v2f* xa = (const v2f*)(X + (long)(GATHER ? gidx[rA] : rA) * D_);
    const v2f* xb = (const v2f*)(X + (long)(GATHER ? gidx[rB] : rB) * D_);

    const v2f* ga = nullptr;
    const v2f* gb = nullptr;
    float sa = 0.0f, sb = 0.0f;
    if (USE_AGG) {
        ga = (const v2f*)(agg + (long)rA * D_);
        gb = (const v2f*)(agg + (long)rB * D_);
        sa = 1.0f / fmaxf(deg[rA], 1.0f);
        sb = 1.0f / fmaxf(deg[rB], 1.0f);
    }

    v8f acc[2][8];
#pragma unroll
    for (int c = 0; c < 8; ++c) {
        float bv = bias[c * 16 + m];
#pragma unroll
        for (int e = 0; e < 8; ++e) { acc[0][c][e] = bv; acc[1][c][e] = bv; }
    }

#pragma unroll 2
    for (int t = 0; t < 32; ++t) {
        int kb = t * 4;
        // A fragments for both row tiles (and both matrices).
        v2f aSa = xa[t * 2 + hi];
        v2f aSb = xb[t * 2 + hi];
        v2f aNa, aNb;
        if (USE_AGG) {
            aNa = ga[t * 2 + hi]; aNa.x *= sa; aNa.y *= sa;
            aNb = gb[t * 2 + hi]; aNb.x *= sb; aNb.y *= sb;
        }
        // Batch all B fragments for this K-step (clause-able, one wait).
        const float* ws0 = W_self + (long)(kb + 2 * hi) * D_ + m;
        const float* wn0 = USE_AGG ? (W_nei + (long)(kb + 2 * hi) * D_ + m) : nullptr;
        v2f bs[8], bn[8];
#pragma unroll
        for (int c = 0; c < 8; ++c) {
            bs[c].x = ws0[c * 16];
            bs[c].y = ws0[c * 16 + D_];
        }
        if (USE_AGG) {
#pragma unroll
            for (int c = 0; c < 8; ++c) {
                bn[c].x = wn0[c * 16];
                bn[c].y = wn0[c * 16 + D_];
            }
        }
        // WMMA burst: each B fragment feeds 2 (or 4) WMMAs.
#pragma unroll
        for (int c = 0; c < 8; ++c) {
            acc[0][c] = __builtin_amdgcn_wmma_f32_16x16x4_f32(
                false, aSa, false, bs[c], (short)0, acc[0][c], false, false);
            acc[1][c] = __builtin_amdgcn_wmma_f32_16x16x4_f32(
                false, aSb, false, bs[c], (short)0, acc[1][c], false, false);
            if (USE_AGG) {
                acc[0][c] = __builtin_amdgcn_wmma_f32_16x16x4_f32(
                    false, aNa, false, bn[c], (short)0, acc[0][c], false, false);
                acc[1][c] = __builtin_amdgcn_wmma_f32_16x16x4_f32(
                    false, aNb, false, bn[c], (short)0, acc[1][c], false, false);
            }
        }
    }

    // C/D layout: VGPR e -> row (16*tb + e + 8*hi), col = 16c + m.
#pragma unroll
    for (int tb = 0; tb < 2; ++tb) {
#pragma unroll
        for (int c = 0; c < 8; ++c) {
#pragma unroll
            for (int e = 0; e < 8; ++e) {
                int orow = row0 + tb * 16 + e + 8 * hi;
                float v = acc[tb][c][e];
                if (ACT) v = tanhf(v);
                if (orow < n_rows)                     // store-only masking
                    out[(long)orow * D_ + c * 16 + m] = v;
            }
        }
    }
}

// out_tail[i] = concat(h1, ent_emb[final_indices], unreach)[order[i]]
__global__ __launch_bounds__(256)
void reorder_kernel(const float* __restrict__ h1,
                    const float* __restrict__ ent_emb,
                    const int*   __restrict__ final_indices,
                    const float* __restrict__ unreach,
                    const int*   __restrict__ order,
                    float*       __restrict__ out_tail)
{
    int i = blockIdx.x * (blockDim.x >> 5) + (threadIdx.x >> 5);
    if (i >= E_) return;
    int lane = threadIdx.x & 31;
    int j = order[i];                       // wave-uniform
    const float4* srow;
    if (j < N1_)              srow = (const float4*)(h1 + (long)j * D_);
    else if (j < N1_ + N2_)   srow = (const float4*)(ent_emb + (long)final_indices[j - N1_] * D_);
    else                      srow = (const float4*)(unreach + (long)(j - N1_ - N2_) * D_);
    ((float4*)(out_tail + (long)i * D_))[lane] = srow[lane];
}

extern "C" void kernel_launch(void* const* d_in, const int* in_sizes, int n_in,
                              void* d_out, int out_size, void* d_ws, size_t ws_size,
                              hipStream_t stream)
{
    const float* ent_emb       = (const float*)d_in[0];
    const float* user_emb      = (const float*)d_in[1];
    const float* rel_emb       = (const float*)d_in[2];
    const float* W_self        = (const float*)d_in[3];
    const float* W_nei         = (const float*)d_in[4];
    const float* bias          = (const float*)d_in[5];
    const int*   final_indices = (const int*)d_in[6];
    const int*   ent_now1      = (const int*)d_in[7];
    const int*   e1_src        = (const int*)d_in[8];
    const int*   e1_dst        = (const int*)d_in[9];
    const int*   e1_rel        = (const int*)d_in[10];
    const int*   e0_src        = (const int*)d_in[11];
    const int*   e0_dst        = (const int*)d_in[12];
    const int*   e0_rel        = (const int*)d_in[13];
    const int*   unreach_idx   = (const int*)d_in[14];
    const int*   order         = (const int*)d_in[15];

    // Workspace layout (floats): [agg1 | deg1 | agg0 | deg0 | h1 | unreach]
    float* ws   = (float*)d_ws;
    float* agg1 = ws;
    float* deg1 = agg1 + (long)N1_ * D_;
    float* agg0 = deg1 + N1_;
    float* deg0 = agg0 + (long)U_ * D_;
    float* h1   = deg0 + U_;
    float* unre = h1 + (long)N1_ * D_;

    long nzero = (long)N1_ * D_ + N1_ + (long)U_ * D_ + U_;
    zero_kernel<<<2048, 256, 0, stream>>>(ws, nzero);

    // hop-1 aggregation: tail = ent_emb gathered through final_indices (== h2)
    edge_scatter_kernel<true><<<(M1_ + 7) / 8, 256, 0, stream>>>(
        ent_emb, final_indices, e1_src, e1_dst, e1_rel, rel_emb, agg1, deg1, M1_);

    // h1 = tanh(ent_emb[ent_now1]@Ws + b + norm(agg1)@Wn)
    {
        int tiles = (N1_ + 31) / 32;                 // 2500
        rgcn_node_kernel<true, true, true><<<(tiles + 7) / 8, 256, 0, stream>>>(
            ent_emb, ent_now1, agg1, deg1, W_self, W_nei, bias, h1, N1_);
    }

    // hop-0 aggregation: tail = h1
    edge_scatter_kernel<false><<<(M0_ + 7) / 8, 256, 0, stream>>>(
        h1, nullptr, e0_src, e0_dst, e0_rel, rel_emb, agg0, deg0, M0_);

    float* out = (float*)d_out;

    // user_rep (no activation) -> first U rows of out
    {
        int tiles = (U_ + 31) / 32;                  // 1563 (last is half tile)
        rgcn_node_kernel<false, true, false><<<(tiles + 7) / 8, 256, 0, stream>>>(
            user_emb, nullptr, agg0, deg0, W_self, W_nei, bias, out, U_);
    }

    // unreachable = tanh(self transform only)
    {
        int tiles = (NU_ + 31) / 32;                 // 625
        rgcn_node_kernel<true, false, true><<<(tiles + 7) / 8, 256, 0, stream>>>(
            ent_emb, unreach_idx, nullptr, nullptr, W_self, W_nei, bias, unre, NU_);
    }

    // out[U + i] = concat(h1, h2, unreach)[order[i]]
    reorder_kernel<<<(E_ + 7) / 8, 256, 0, stream>>>(
        h1, ent_emb, final_indices, unre, order, out + (long)U_ * D_);
}